// MithralNN_23390391894939
// MI455X (gfx1250) — compile-verified
//
#include <hip/hip_runtime.h>

// Mithral PQ approximate matmul for MI455X (gfx1250, wave32).
// encode: A = (-2*protos) from LDS, B = X^T streamed, C init = ||p||^2
//         -> V_WMMA_F32_16X16X4_F32 chain emits distances directly.
//         Per-lane branchless 8-way min + one cross-half shfl = argmin.
// decode: scalar nibble-code fetch + L2-resident LUT gathers, NT stores.

#define N_ROWS 32768
#define DIM    512
#define CB     16    // codebooks
#define KW     16    // codewords per codebook
#define SUB    32    // subvector length
#define MOUT   1024  // output columns
#define WPB    4     // waves per block in encode

typedef float v2f __attribute__((ext_vector_type(2)));
typedef float v4f __attribute__((ext_vector_type(4)));
typedef float v8f __attribute__((ext_vector_type(8)));

// Monotone map of f32 bits to unsigned order, packed with the codeword index
// so u64-min == (min dist, then min index) — matches jnp.argmin tie-break.
__device__ __forceinline__ unsigned long long packkey(float d, int idx) {
    unsigned u = __float_as_uint(d);
    u ^= ((unsigned)((int)u >> 31)) | 0x80000000u;
    return ((unsigned long long)u << 32) | (unsigned)idx;
}

// ---------------------------------------------------------------------------
// Kernel 1: PQ encode. One wave per 16-row tile of X.
//   A 16x4 frag (M=codeword): -2*protos, staged once per block in LDS.
//   B 4x16 frag (N=row):      X^T, streamed from global (read exactly once;
//                             also feeds the fused mean(X) partial sum).
//   C/D: dist[kw][row]; VGPR j / lane l<16 -> kw=j, row=l; lanes 16-31 -> kw=j+8.
// ---------------------------------------------------------------------------
__global__ __launch_bounds__(32 * WPB) void mithral_encode(
    const float* __restrict__ X, const float* __restrict__ protos,
    unsigned long long* __restrict__ codes64, float* __restrict__ partials)
{
    __shared__ float sP[CB * KW * SUB];   // -2 * protos  (32 KB)
    __shared__ float sPn[CB * KW];        // ||proto||^2  (1 KB)
    __shared__ float wavesum[WPB];

    const int tid = threadIdx.x;

    // Stage -2*protos into LDS (vectorized, coalesced; all 4 waves reuse it).
    {
        const v4f* src = (const v4f*)protos;
        v4f*       dst = (v4f*)sP;
        #pragma unroll
        for (int i = tid; i < CB * KW * SUB / 4; i += 32 * WPB) {
            v4f v = src[i];
            dst[i] = v * -2.f;
        }
    }
    // ||p||^2 per codeword.
    for (int i = tid; i < CB * KW; i += 32 * WPB) {
        const float* p = protos + (size_t)i * SUB;
        float s = 0.f;
        #pragma unroll
        for (int j = 0; j < SUB; ++j) s += p[j] * p[j];
        sPn[i] = s;
    }
    __syncthreads();

    const int wave = tid >> 5;
    const int lane = tid & 31;
    const int half = lane >> 4;   // 0: lanes 0-15, 1: lanes 16-31
    const int l15  = lane & 15;
    const int row0 = (blockIdx.x * WPB + wave) * 16;

    float asum = 0.f;
    unsigned long long pk = 0ull;   // nibble-packed codes for row row0+l15

    for (int c = 0; c < CB; ++c) {
        // C init = ||p||^2 for kw = j + half*8 (same for all lanes in half).
        const v4f pna = *(const v4f*)(sPn + c * KW + half * 8);
        const v4f pnb = *(const v4f*)(sPn + c * KW + half * 8 + 4);
        v8f acc;
        acc[0] = pna.x; acc[1] = pna.y; acc[2] = pna.z; acc[3] = pna.w;
        acc[4] = pnb.x; acc[5] = pnb.y; acc[6] = pnb.z; acc[7] = pnb.w;

        #pragma unroll
        for (int s0 = 0; s0 < SUB; s0 += 4) {
            const int off = s0 + half * 2;
            // A frag from LDS: -2*protos[c][kw=l15][off..off+1]
            const v2f a = *(const v2f*)(sP + ((size_t)c * KW + l15) * SUB + off);
            // B frag from global: X[row0+l15][c*SUB+off..+1] (each element once)
            const v2f b = *(const v2f*)(X + (size_t)(row0 + l15) * DIM + c * SUB + off);
            asum += b.x + b.y;
            acc = __builtin_amdgcn_wmma_f32_16x16x4_f32(
                false, a, false, b, (short)0, acc, false, false);
        }

        // Per-lane branchless min over 8 codewords (kw = half*8 + j).
        unsigned long long best = packkey(acc[0], half * 8);
        #pragma unroll
        for (int j = 1; j < 8; ++j) {
            const unsigned long long k = packkey(acc[j], half * 8 + j);
            best = (k < best) ? k : best;
        }
        // Combine halves: lanes l and l+16 both end with argmin for row l15.
        const unsigned long long ob = __shfl_xor(best, 16, 32);
        best = (ob < best) ? ob : best;

        pk |= (best & 15ull) << (4 * c);
    }

    if (half == 0) codes64[row0 + l15] = pk;   // one b64 store per row

    // Deterministic tree reduction of the X partial sum.
    #pragma unroll
    for (int m = 1; m < 32; m <<= 1) asum += __shfl_xor(asum, m, 32);
    if (lane == 0) wavesum[wave] = asum;
    __syncthreads();
    if (tid == 0) {
        float s = 0.f;
        #pragma unroll
        for (int w = 0; w < WPB; ++w) s += wavesum[w];
        partials[blockIdx.x] = s;
    }
}

// ---------------------------------------------------------------------------
// Kernel 2: deterministic reduction of 512 per-block partials -> mean(X).
// ---------------------------------------------------------------------------
__global__ __launch_bounds__(256) void mithral_reduce(
    const float* __restrict__ partials, float* __restrict__ mean)
{
    __shared__ float sh[256];
    const int t = threadIdx.x;
    sh[t] = partials[t] + partials[t + 256];
    __syncthreads();
    for (int m = 128; m > 0; m >>= 1) {
        if (t < m) sh[t] += sh[t + m];
        __syncthreads();
    }
    if (t == 0) mean[0] = sh[0] * (1.f / ((float)N_ROWS * (float)DIM));
}

// ---------------------------------------------------------------------------
// Kernel 3: decode. One block per row; 256 threads x v4f = 1024 columns.
// Codes arrive as one scalar u64 (SALU extract, no LDS/sync); 16 gathers per
// thread from the 1 MB L2-resident luts; non-temporal coalesced store.
// ---------------------------------------------------------------------------
__global__ __launch_bounds__(256) void mithral_decode(
    const unsigned long long* __restrict__ codes64, const float* __restrict__ luts,
    const float* __restrict__ mean, float* __restrict__ out)
{
    const int n = blockIdx.x;
    const int t = threadIdx.x;
    const unsigned long long pk = codes64[n];   // uniform -> scalar load
    const float scale = mean[0];

    v4f acc = {0.f, 0.f, 0.f, 0.f};
    #pragma unroll
    for (int c = 0; c < CB; ++c) {
        const int code = (int)((pk >> (4 * c)) & 15ull);
        const v4f v = ((const v4f*)(luts + (size_t)(c * KW + code) * MOUT))[t];
        acc += v;
    }
    const v4f r = acc * scale;
    __builtin_nontemporal_store(r, (v4f*)(out + (size_t)n * MOUT) + t);
}

// ---------------------------------------------------------------------------
extern "C" void kernel_launch(void* const* d_in, const int* in_sizes, int n_in,
                              void* d_out, int out_size, void* d_ws, size_t ws_size,
                              hipStream_t stream)
{
    (void)in_sizes; (void)n_in; (void)out_size; (void)ws_size;

    const float* X      = (const float*)d_in[0];   // [N, D] f32
    const float* protos = (const float*)d_in[1];   // [C, K, S] f32
    const float* luts   = (const float*)d_in[2];   // [C, K, M] f32
    float*       out    = (float*)d_out;           // [N, M] f32

    // Scratch: [0,4) mean | [256, +2KB) partials | [4096, +N*8) packed codes
    float*              mean     = (float*)d_ws;
    float*              partials = (float*)((char*)d_ws + 256);
    unsigned long long* codes64  = (unsigned long long*)((char*)d_ws + 4096);

    const int eblks = (N_ROWS / 16) / WPB;   // 512

    mithral_encode<<<eblks, 32 * WPB, 0, stream>>>(X, protos, codes64, partials);
    mithral_reduce<<<1, 256, 0, stream>>>(partials, mean);
    mithral_decode<<<N_ROWS, 256, 0, stream>>>(codes64, luts, mean, out);
}